// FourDirGradientConv_28089086116538
// MI455X (gfx1250) — compile-verified
//
#include <hip/hip_runtime.h>

typedef __attribute__((ext_vector_type(2))) float v2f;
typedef __attribute__((ext_vector_type(8))) float v8f;

#define B_N 8
#define C_N 32
#define H_N 512
#define W_N 512
#define TH 8       // output rows per block
#define TW 128     // output cols per block
#define NT 512     // threads per block (16 waves)
#define LROW 140   // LDS row stride in floats (136 stored + pad, 16B aligned)
#define NVEC 340   // 10 rows * 34 float4 per row

// LDS-only workgroup barrier: does NOT flush outstanding global loads.
#define WG_LDS_BARRIER()                                  \
    asm volatile("s_wait_dscnt 0x0\n\t"                   \
                 "s_barrier_signal -1\n\t"                \
                 "s_barrier_wait -1" ::: "memory")

// ---------------------------------------------------------------------------
// Pass 0: zero the stats accumulators (ws[0..3]=sum, ws[4..7]=sumsq)
// ---------------------------------------------------------------------------
__global__ void zero_stats_kernel(float* ws) {
    if (threadIdx.x < 8) ws[threadIdx.x] = 0.0f;
}

// ---------------------------------------------------------------------------
// Pass 1: fused 4-dir stencil + 1x1 conv (WMMA f32 16x16x4) + stat accumulation
//   512 threads = 16 waves; wave wv owns half of output row h0+(wv>>1).
//   Distance-2 software pipeline: channel c+2's load is issued while channel
//   c is computed, consumed (committed to LDS) two barrier regions later.
// ---------------------------------------------------------------------------
__global__ __launch_bounds__(NT) void fourdir_wmma_kernel(
    const float* __restrict__ x,     // [8,32,512,512]
    const float* __restrict__ w,     // [4,128] (k = dir*32 + c)
    float* __restrict__ y,           // [8,4,512,512] (un-normalized, in d_out)
    float* __restrict__ stats)       // [8] float atomics in d_ws
{
    __shared__ float xs[2][10 * LROW];  // double-buffered 10 x 136 halo tile
    __shared__ float wl[16 * 128];      // weights padded to 16 rows (rows 4..15 = 0)
    __shared__ float red[16 * 8];
    __shared__ float dump[4];           // sink for boundary-OOB lane stores

    const int tid  = threadIdx.x;
    const int lane = tid & 31;
    const int wv   = tid >> 5;           // 0..15
    const int row  = wv >> 1;            // 0..7  : output row within tile
    const int colh = (wv & 1) * 64;      // 0/64  : column half within tile

    const int bi = blockIdx.x;
    const int wt = bi & 3;
    const int ht = (bi >> 2) & 63;
    const int b  = bi >> 8;
    const int h0 = ht * TH;
    const int w0 = wt * TW;

    // padded weights (rows >=4 exactly zero) + pre-zero both halo buffers
    for (int i = tid; i < 16 * 128; i += NT) wl[i] = (i < 512) ? w[i] : 0.0f;
    for (int i = tid; i < 2 * 10 * LROW; i += NT) ((float*)xs)[i] = 0.0f;

    // ---- loop-invariant staging slot (1 float4 per thread) ----
    // vector v (0..339): row r = v/34, jv = v%34, covers cols [w0-4, w0+132)
    const bool act0 = (tid < NVEC);
    const int  r0 = tid / 34, j0 = tid - r0 * 34;
    const int  gh0 = h0 - 1 + r0, gw0 = w0 - 4 + j0 * 4;
    const bool ok0 = act0 && ((unsigned)gh0 < (unsigned)H_N) && ((unsigned)gw0 < (unsigned)W_N);
    const int  goff0 = ok0 ? (gh0 * W_N + gw0) : 0;   // clamped, always in-bounds
    const int  loff  = (act0 ? (r0 * LROW + j0 * 4) : 0);
    // destination pointers: real slot, or dump sink for boundary-OOB lanes
    float* const dst0 = ok0 ? &xs[0][loff] : dump;
    float* const dst1 = ok0 ? &xs[1][loff] : dump;

    const size_t planeStride = (size_t)H_N * W_N;
    const float* xbase = x + (size_t)b * C_N * planeStride;

    auto stage = [&](int c, float4& s0) {
        s0 = *(const float4*)(xbase + (size_t)c * planeStride + goff0);  // raw
    };
    auto commit = [&](float* dst, const float4& s0) {
        if (act0) *(float4*)dst = s0;    // single store, no select
    };

    __syncthreads();   // wl + xs zero-init visible before first commit

    v8f acc[4] = {};
    const int mrow = lane & 15;
    const int half = lane >> 4;
    const int nrow = (lane < 16) ? row : (row + 2);

    auto compute = [&](int c, int buf) {
        const float* xsb = &xs[buf][0];
        v2f a;
        a.x = wl[mrow * 128 + (half * 2 + 0) * C_N + c];
        a.y = wl[mrow * 128 + (half * 2 + 1) * C_N + c];
#pragma unroll
        for (int t = 0; t < 4; ++t) {
            const int col = 4 + colh + t * 16 + mrow;
            const float center = xsb[(row + 1) * LROW + col];
            const float vp     = xsb[nrow * LROW + col + 1];
            const float vm     = xsb[nrow * LROW + col - 1];
            v2f bf;
            bf.x = vp - center;   // ne / se
            bf.y = vm - center;   // nw / sw
            acc[t] = __builtin_amdgcn_wmma_f32_16x16x4_f32(
                false, a, false, bf, (short)0, acc[t], false, false);
        }
    };

    // ---- distance-2 software pipeline: 1 LDS-only barrier per channel ----
    float4 sA, sB;
    stage(0, sA);
    stage(1, sB);
    for (int c = 0; c < C_N; c += 2) {
        commit(dst0, sA);                  // waits for load issued 2 regions ago
        if (c + 2 < C_N) stage(c + 2, sA); // in flight across 2 barrier regions
        WG_LDS_BARRIER();
        compute(c, 0);

        commit(dst1, sB);
        if (c + 3 < C_N) stage(c + 3, sB);
        WG_LDS_BARRIER();
        compute(c + 1, 1);
    }

    // ---- per-channel sum / sumsq (lanes 16-31 hold exact zeros) ----
    float s[4] = {0, 0, 0, 0}, q[4] = {0, 0, 0, 0};
#pragma unroll
    for (int t = 0; t < 4; ++t)
#pragma unroll
        for (int o = 0; o < 4; ++o) {
            float v = acc[t][o];
            s[o] += v;
            q[o] += v * v;
        }

    // ---- store y: D VGPR o, lanes 0-15 = channel o, pixel = lane ----
    if (lane < 16) {
        const int gh = h0 + row;
#pragma unroll
        for (int t = 0; t < 4; ++t) {
            const int gw = w0 + colh + t * 16 + lane;
#pragma unroll
            for (int o = 0; o < 4; ++o)
                y[(((size_t)(b * 4 + o)) * H_N + gh) * W_N + gw] = acc[t][o];
        }
    }

    // ---- block reduction + global atomics ----
#pragma unroll
    for (int off = 16; off >= 1; off >>= 1)
#pragma unroll
        for (int o = 0; o < 4; ++o) {
            s[o] += __shfl_xor(s[o], off, 32);
            q[o] += __shfl_xor(q[o], off, 32);
        }
    if (lane == 0)
#pragma unroll
        for (int o = 0; o < 4; ++o) {
            red[wv * 8 + o]     = s[o];
            red[wv * 8 + 4 + o] = q[o];
        }
    __syncthreads();
    if (tid < 8) {
        float t = 0.0f;
        for (int v = 0; v < 16; ++v) t += red[v * 8 + tid];
        atomicAdd(stats + tid, t);
    }
}

// ---------------------------------------------------------------------------
// Pass 2: stats -> per-channel scale/shift (ws[8+2o]=scale, ws[9+2o]=shift)
// ---------------------------------------------------------------------------
__global__ void bn_coef_kernel(const float* __restrict__ stats,
                               const float* __restrict__ gamma,
                               const float* __restrict__ beta,
                               float* __restrict__ coef) {
    const int o = threadIdx.x;
    if (o < 4) {
        const float n    = (float)B_N * (float)H_N * (float)W_N;
        const float mean = stats[o] / n;
        const float var  = stats[4 + o] / n - mean * mean;
        const float sc   = gamma[o] * rsqrtf(var + 1e-5f);
        coef[8 + 2 * o] = sc;
        coef[9 + 2 * o] = beta[o] - mean * sc;
    }
}

// ---------------------------------------------------------------------------
// Pass 3: in-place normalize y (float4 vectorized; plane = 65536 float4s)
// ---------------------------------------------------------------------------
__global__ __launch_bounds__(256) void bn_apply_kernel(float4* __restrict__ y,
                                                       const float* __restrict__ ws) {
    const size_t i = (size_t)blockIdx.x * 256 + threadIdx.x;
    const int o = (int)((i >> 16) & 3);
    const float sc = ws[8 + 2 * o];
    const float sh = ws[9 + 2 * o];
    float4 v = y[i];
    v.x = fmaf(v.x, sc, sh);
    v.y = fmaf(v.y, sc, sh);
    v.z = fmaf(v.z, sc, sh);
    v.w = fmaf(v.w, sc, sh);
    y[i] = v;
}

extern "C" void kernel_launch(void* const* d_in, const int* in_sizes, int n_in,
                              void* d_out, int out_size, void* d_ws, size_t ws_size,
                              hipStream_t stream) {
    const float* x     = (const float*)d_in[0];
    const float* w     = (const float*)d_in[1];
    const float* gamma = (const float*)d_in[2];
    const float* beta  = (const float*)d_in[3];
    float* out = (float*)d_out;
    float* ws  = (float*)d_ws;

    zero_stats_kernel<<<1, 32, 0, stream>>>(ws);
    fourdir_wmma_kernel<<<B_N * (H_N / TH) * (W_N / TW), NT, 0, stream>>>(x, w, out, ws);
    bn_coef_kernel<<<1, 4, 0, stream>>>(ws, gamma, beta, ws);
    const int n4 = B_N * 4 * H_N * W_N / 4;
    bn_apply_kernel<<<n4 / 256, 256, 0, stream>>>((float4*)out, ws);
}